// InputRefXGMSoftmaxMulEpsilonModuleVMAP_23381801959832
// MI455X (gfx1250) — compile-verified
//
#include <hip/hip_runtime.h>
#include <stdint.h>

// softmax(x, axis=-1) * v
//   x   : [B=4, H=16, S=1024, S=1024] f32  (65536 rows of 1024)
//   v   : [B, H, 1, S] f32                 (one 1024-vector per (b,h))
//   out : same shape as x, f32
//
// Memory-bound: ~512 MB of traffic -> ~22 us floor at 23.3 TB/s.
// Strategy: one wave32 per row, whole row in registers, NT streaming
// loads/stores, v[bh,:] staged once per block into LDS via the CDNA5
// async global->LDS path (GLOBAL_LOAD_ASYNC_TO_LDS_B128 + s_wait_asynccnt).

typedef float v4f __attribute__((ext_vector_type(4)));
typedef int   v4i __attribute__((vector_size(16)));   // matches builtin param type

#define SEQ 1024
#define WAVES_PER_BLOCK 8
#define THREADS (WAVES_PER_BLOCK * 32)
#define VPL 8 /* float4 per lane: 1024 / 32 lanes / 4 */

#if defined(__HIP_DEVICE_COMPILE__) &&                                   \
    __has_builtin(__builtin_amdgcn_global_load_async_to_lds_b128) &&     \
    __has_builtin(__builtin_amdgcn_s_wait_asynccnt)
#define USE_ASYNC_LDS 1
#else
#define USE_ASYNC_LDS 0
#endif

__global__ __launch_bounds__(THREADS)
void softmax_mul_rows(const float* __restrict__ x,
                      const float* __restrict__ v,
                      float* __restrict__ out,
                      long long nrows)
{
    __shared__ float vs[SEQ];

    const int lane = threadIdx.x & 31;
    const int wave = threadIdx.x >> 5;
    const long long row0 = (long long)blockIdx.x * WAVES_PER_BLOCK;
    const long long bh   = row0 >> 10; // 8 | 1024 -> whole block shares (b,h)

    // ---- stage v[bh, :] (4 KB) into LDS: one float4 per thread ----
    {
        const float* gsrc = v + bh * SEQ + (size_t)threadIdx.x * 4;
        float*       ldst = &vs[threadIdx.x * 4];
#if USE_ASYNC_LDS
        // CDNA5 async copy: GLOBAL_LOAD_ASYNC_TO_LDS_B128 (ASYNCcnt-tracked).
        // Toolchain signature (from round-1 diagnostic): generic v4i* params.
        __builtin_amdgcn_global_load_async_to_lds_b128(
            (v4i*)gsrc, (v4i*)ldst, /*offset=*/0, /*cpol=*/0);
        __builtin_amdgcn_s_wait_asynccnt(0);
#else
        *(v4f*)ldst = *(const v4f*)gsrc;
#endif
    }
    __syncthreads(); // each wave waited its own async cnt before arriving

    const long long row = row0 + wave;
    if (row >= nrows) return;

    const v4f* xr   = (const v4f*)(x   + row * SEQ);
    v4f*       outr = (v4f*)      (out + row * SEQ);

    // ---- streaming (non-temporal) load of the whole row into registers ----
    v4f d[VPL];
#pragma unroll
    for (int j = 0; j < VPL; ++j)
        d[j] = __builtin_nontemporal_load(xr + j * 32 + lane);

    // ---- wave-wide max (wave32: 5 xor-shuffle steps) ----
    float m = -3.402823466e+38f;
#pragma unroll
    for (int j = 0; j < VPL; ++j)
        m = fmaxf(m, fmaxf(fmaxf(d[j][0], d[j][1]), fmaxf(d[j][2], d[j][3])));
#pragma unroll
    for (int off = 16; off > 0; off >>= 1)
        m = fmaxf(m, __shfl_xor(m, off, 32));

    // ---- exp(x - max) + wave-wide sum ----
    float s = 0.0f;
#pragma unroll
    for (int j = 0; j < VPL; ++j) {
        d[j][0] = __expf(d[j][0] - m);
        d[j][1] = __expf(d[j][1] - m);
        d[j][2] = __expf(d[j][2] - m);
        d[j][3] = __expf(d[j][3] - m);
        s += (d[j][0] + d[j][1]) + (d[j][2] + d[j][3]);
    }
#pragma unroll
    for (int off = 16; off > 0; off >>= 1)
        s += __shfl_xor(s, off, 32);
    const float inv = 1.0f / s;

    // ---- scale by 1/sum and by v (from LDS), stream out non-temporally ----
#pragma unroll
    for (int j = 0; j < VPL; ++j) {
        v4f vv = *(const v4f*)&vs[(j * 32 + lane) * 4];
        v4f r  = d[j] * inv * vv;
        __builtin_nontemporal_store(r, outr + j * 32 + lane);
    }
}

extern "C" void kernel_launch(void* const* d_in, const int* in_sizes, int n_in,
                              void* d_out, int out_size, void* d_ws, size_t ws_size,
                              hipStream_t stream)
{
    (void)n_in; (void)out_size; (void)d_ws; (void)ws_size;
    const float* x = (const float*)d_in[0];
    const float* v = (const float*)d_in[1];
    float*       o = (float*)d_out;

    const long long nrows  = (long long)in_sizes[0] / SEQ; // 65536
    const int       blocks = (int)((nrows + WAVES_PER_BLOCK - 1) / WAVES_PER_BLOCK);

    softmax_mul_rows<<<blocks, THREADS, 0, stream>>>(x, v, o, nrows);
}